// MuxSimpleGNN_58385785421917
// MI455X (gfx1250) — compile-verified
//
#include <hip/hip_runtime.h>
#include <hip/hip_bf16.h>

typedef float v2f __attribute__((ext_vector_type(2)));
typedef float v8f __attribute__((ext_vector_type(8)));

#define HDIM 64
#define LRELU_SLOPE 0.2f

// ---------- order-preserving float<->int encoding for atomicMax ----------
__device__ __forceinline__ int enc_f(float f) {
    int i = __float_as_int(f);
    return i >= 0 ? i : (i ^ 0x7FFFFFFF);
}
__device__ __forceinline__ float dec_f(int i) {
    return __int_as_float(i >= 0 ? i : (i ^ 0x7FFFFFFF));
}

// ---------- WMMA GEMM:  Y[M,64] = act(X[M,K] @ W[K,64] (+bias)) ----------
// block = 128 threads (4 waves); wave w handles 16-col tile; blockIdx = 16-row tile.
// V_WMMA_F32_16X16X4_F32 fragment layout (ISA 7.12.2):
//   A 16x4:  lanes 0-15 rows 0-15 hold K={k0,k0+1}; lanes 16-31 hold K={k0+2,k0+3}
//   B 4x16:  lanes 0-15 cols 0-15 hold K={k0,k0+1}; lanes 16-31 hold K={k0+2,k0+3}
//   C/D:     VGPR i -> row i + 8*(lane>=16), col = lane&15
template <int K, bool RELU, bool BIAS>
__global__ void k_gemm_wmma(const float* __restrict__ X, const float* __restrict__ W,
                            const float* __restrict__ bias, float* __restrict__ Y, int M) {
    const int wave = threadIdx.x >> 5;
    const int lane = threadIdx.x & 31;
    const int half = lane >> 4;      // 0 or 1
    const int l15  = lane & 15;
    const int row0 = blockIdx.x * 16;
    const int col0 = wave * 16;
    const bool full = (row0 + 16 <= M);   // uniform across wave
    int arow = row0 + l15;
    if (!full && arow >= M) arow = M - 1; // clamp OOB reads only on ragged tile
    const int bcol = col0 + l15;

    v8f acc = {};
#pragma unroll
    for (int k0 = 0; k0 < K; k0 += 4) {
        const int ka = k0 + half * 2;
        v2f a, b;
        a.x = X[(size_t)arow * K + ka];
        a.y = X[(size_t)arow * K + ka + 1];
        b.x = W[(size_t)ka * HDIM + bcol];
        b.y = W[(size_t)(ka + 1) * HDIM + bcol];
        acc = __builtin_amdgcn_wmma_f32_16x16x4_f32(false, a, false, b,
                                                    (short)0, acc, false, false);
    }

    const float bv = BIAS ? bias[bcol] : 0.0f;
    if (full) {
        // straight-line epilogue: no divergence, stores can clause
#pragma unroll
        for (int vi = 0; vi < 8; ++vi) {
            const int r = row0 + vi + 8 * half;
            float v = acc[vi];
            if (BIAS) v += bv;
            if (RELU) v = fmaxf(v, 0.0f);
            Y[(size_t)r * HDIM + bcol] = v;
        }
    } else {
#pragma unroll
        for (int vi = 0; vi < 8; ++vi) {
            const int r = row0 + vi + 8 * half;
            if (r < M) {
                float v = acc[vi];
                if (BIAS) v += bv;
                if (RELU) v = fmaxf(v, 0.0f);
                Y[(size_t)r * HDIM + bcol] = v;
            }
        }
    }
}

// ---------- per-node: attention logits e_s/e_d + init m/s/msg ----------
__global__ void k_node_scores(const float* __restrict__ h,
                              const float* __restrict__ a_src,
                              const float* __restrict__ a_dst,
                              float* __restrict__ es, float* __restrict__ ed,
                              int* __restrict__ mEnc, float* __restrict__ s,
                              float* __restrict__ msg, int n) {
    const int i = blockIdx.x * blockDim.x + threadIdx.x;
    if (i >= n) return;
    const float* hr = h + (size_t)i * HDIM;
    float s1 = 0.f, s2 = 0.f;
#pragma unroll
    for (int c = 0; c < HDIM; ++c) {
        const float v = hr[c];
        s1 += v * a_src[c];
        s2 += v * a_dst[c];
    }
    es[i] = s1;
    ed[i] = s2;
    mEnc[i] = (int)0x80000000;  // "-inf" sentinel (never decoded if no in-edges)
    s[i] = 0.f;
    float* mr = msg + (size_t)i * HDIM;
#pragma unroll
    for (int c = 0; c < HDIM; ++c) mr[c] = 0.f;
}

// ---------- per-edge pass 1: e = leaky_relu(es[src]+ed[dst]); segment max ----------
__global__ void k_edge_scores(const int* __restrict__ src, const int* __restrict__ dst,
                              const float* __restrict__ es, const float* __restrict__ ed,
                              float* __restrict__ ebuf, int* __restrict__ mEnc, int E) {
    const int i = blockIdx.x * blockDim.x + threadIdx.x;
    if (i >= E) return;
    const int u = src[i], v = dst[i];
    float e = es[u] + ed[v];
    e = e > 0.f ? e : LRELU_SLOPE * e;
    ebuf[i] = e;
    atomicMax(&mEnc[v], enc_f(e));
}

// ---------- per-edge pass 2: w = exp(e - m[dst]); segment sum ----------
__global__ void k_edge_expsum(const int* __restrict__ dst, float* __restrict__ ebuf,
                              const int* __restrict__ mEnc, float* __restrict__ s, int E) {
    const int i = blockIdx.x * blockDim.x + threadIdx.x;
    if (i >= E) return;
    const int v = dst[i];
    const float w = __expf(ebuf[i] - dec_f(mEnc[v]));
    ebuf[i] = w;  // overwrite e with w
    atomicAdd(&s[v], w);
}

// ---------- per-edge pass 3: msg[dst] += h[src] * w  (64 lanes per edge) ----------
// block = 256 threads = 4 edges; lane 0 of each edge-group loads (src,dst,w) once
// into LDS; all 64 lanes of the group read the broadcast.
__global__ void k_edge_scatter(const int* __restrict__ src, const int* __restrict__ dst,
                               const float* __restrict__ w, const float* __restrict__ h,
                               float* __restrict__ msg, int E) {
    __shared__ int   sh_u[4];
    __shared__ int   sh_v[4];
    __shared__ float sh_w[4];
    const int grp = threadIdx.x >> 6;          // 0..3
    const int c   = threadIdx.x & 63;          // channel
    const int e   = blockIdx.x * 4 + grp;
    if (c == 0 && e < E) {
        sh_u[grp] = src[e];
        sh_v[grp] = dst[e];
        sh_w[grp] = w[e];
    }
    __syncthreads();
    if (e >= E) return;
    const int u = sh_u[grp], v = sh_v[grp];
    const float we = sh_w[grp];
    atomicAdd(&msg[(size_t)v * HDIM + c], h[(size_t)u * HDIM + c] * we);
}

// ---------- per-node fusion: normalize msgs, gate-softmax, residual+LN+relu ----------
// one wave32 per node, each lane owns channels (lane, lane+32)
__global__ void k_fuse_ln(const float* __restrict__ x_in,
                          const float* __restrict__ msg0, const float* __restrict__ s0,
                          const float* __restrict__ msg1, const float* __restrict__ s1,
                          const float* __restrict__ gat_b,   // [2][64]
                          const float* __restrict__ gate_w,  // [64][2]
                          const float* __restrict__ gate_b,  // [2]
                          const float* __restrict__ ln_g, const float* __restrict__ ln_b,
                          float* __restrict__ x_out, int n) {
    const int lane = threadIdx.x & 31;
    const int node = blockIdx.x * 8 + (threadIdx.x >> 5);
    if (node >= n) return;
    const int c0 = lane, c1 = lane + 32;
    const size_t base = (size_t)node * HDIM;
    const float xv0 = x_in[base + c0], xv1 = x_in[base + c1];

    // gate logits (wave reduction over channels)
    float g0p = xv0 * gate_w[c0 * 2 + 0] + xv1 * gate_w[c1 * 2 + 0];
    float g1p = xv0 * gate_w[c0 * 2 + 1] + xv1 * gate_w[c1 * 2 + 1];
#pragma unroll
    for (int m = 16; m >= 1; m >>= 1) {
        g0p += __shfl_xor(g0p, m, 32);
        g1p += __shfl_xor(g1p, m, 32);
    }
    const float gl0 = g0p + gate_b[0], gl1 = g1p + gate_b[1];
    const float mx = fmaxf(gl0, gl1);
    const float e0 = __expf(gl0 - mx), e1 = __expf(gl1 - mx);
    const float ginv = 1.f / (e0 + e1);
    const float g0 = e0 * ginv, g1 = e1 * ginv;

    // normalize per-relation messages, add bias, fuse
    const float i0 = 1.f / (s0[node] + 1e-16f);
    const float i1 = 1.f / (s1[node] + 1e-16f);
    const float r0c0 = msg0[base + c0] * i0 + gat_b[c0];
    const float r0c1 = msg0[base + c1] * i0 + gat_b[c1];
    const float r1c0 = msg1[base + c0] * i1 + gat_b[HDIM + c0];
    const float r1c1 = msg1[base + c1] * i1 + gat_b[HDIM + c1];
    const float y0 = xv0 + g0 * r0c0 + g1 * r1c0;
    const float y1 = xv1 + g0 * r0c1 + g1 * r1c1;

    // layernorm over 64 channels
    float sm = y0 + y1, sq = y0 * y0 + y1 * y1;
#pragma unroll
    for (int m = 16; m >= 1; m >>= 1) {
        sm += __shfl_xor(sm, m, 32);
        sq += __shfl_xor(sq, m, 32);
    }
    const float mean = sm * (1.f / 64.f);
    const float var = sq * (1.f / 64.f) - mean * mean;
    const float rstd = rsqrtf(var + 1e-5f);
    const float o0 = (y0 - mean) * rstd * ln_g[c0] + ln_b[c0];
    const float o1 = (y1 - mean) * rstd * ln_g[c1] + ln_b[c1];
    x_out[base + c0] = fmaxf(o0, 0.f);
    x_out[base + c1] = fmaxf(o1, 0.f);
}

// ---------- pooling ----------
__global__ void k_zero64(float* __restrict__ p) { p[threadIdx.x] = 0.f; }

__global__ void k_pool_partial(const float* __restrict__ x, float* __restrict__ sum, int n) {
    const int c = threadIdx.x;  // 64 threads
    float acc = 0.f;
    for (int i = blockIdx.x; i < n; i += gridDim.x) acc += x[(size_t)i * HDIM + c];
    atomicAdd(&sum[c], acc);
}

__global__ void k_pool_final(const float* __restrict__ sum, const float* __restrict__ pool_w,
                             const float* __restrict__ pool_b, float* __restrict__ out, int n) {
    const int c = threadIdx.x;  // 64 threads
    const float invn = 1.f / (float)n;
    float acc = 0.f;
#pragma unroll
    for (int k = 0; k < HDIM; ++k) acc += (sum[k] * invn) * pool_w[k * HDIM + c];
    out[c] = acc + pool_b[c];
}

extern "C" void kernel_launch(void* const* d_in, const int* in_sizes, int n_in,
                              void* d_out, int out_size, void* d_ws, size_t ws_size,
                              hipStream_t stream) {
    const int FD = 16, H = HDIM, L = 2;
    const int N = in_sizes[0] / FD;   // 50000
    const int E = in_sizes[1] / 2;    // 800000

    const float* nf     = (const float*)d_in[0];
    const int*   ep     = (const int*)d_in[1];
    const int*   em     = (const int*)d_in[2];
    const float* in_w   = (const float*)d_in[3];
    const float* in_b   = (const float*)d_in[4];
    const float* gat_w  = (const float*)d_in[5];   // [L,2,64,64]
    const float* a_src  = (const float*)d_in[6];   // [L,2,1,64]
    const float* a_dst  = (const float*)d_in[7];
    const float* gat_b  = (const float*)d_in[8];   // [L,2,64]
    const float* gate_w = (const float*)d_in[9];   // [L,64,2]
    const float* gate_b = (const float*)d_in[10];  // [L,2]
    const float* ln_g   = (const float*)d_in[11];  // [L,64]
    const float* ln_b   = (const float*)d_in[12];
    const float* pool_w = (const float*)d_in[13];  // [64,64]
    const float* pool_b = (const float*)d_in[14];

    float* ws = (float*)d_ws;
    size_t o = 0;
    float* x0    = ws + o; o += (size_t)N * H;
    float* x1    = ws + o; o += (size_t)N * H;
    float* h     = ws + o; o += (size_t)N * H;
    float* msg0  = ws + o; o += (size_t)N * H;
    float* msg1  = ws + o; o += (size_t)N * H;
    float* es    = ws + o; o += N;
    float* ed    = ws + o; o += N;
    float* s0    = ws + o; o += N;
    float* s1    = ws + o; o += N;
    int*   mEnc  = (int*)(ws + o); o += N;
    float* ebuf  = ws + o; o += E;
    float* sum64 = ws + o; o += 64;

    float* out_nodes = (float*)d_out;
    float* out_graph = out_nodes + (size_t)N * H;

    const int rowTiles = (N + 15) / 16;

    // x0 = relu(nf @ in_w + in_b)   [WMMA, K=16]
    k_gemm_wmma<16, true, true><<<rowTiles, 128, 0, stream>>>(nf, in_w, in_b, x0, N);

    const int* srcs[2] = {ep, em};
    for (int l = 0; l < L; ++l) {
        const float* xin  = (l == 0) ? x0 : x1;
        float*       xout = (l == 0) ? x1 : out_nodes;
        for (int r = 0; r < 2; ++r) {
            const float* W   = gat_w + (size_t)(l * 2 + r) * H * H;
            const float* as_ = a_src + (size_t)(l * 2 + r) * H;
            const float* ad_ = a_dst + (size_t)(l * 2 + r) * H;
            float* msg = (r == 0) ? msg0 : msg1;
            float* s   = (r == 0) ? s0 : s1;
            const int* src = srcs[r];
            const int* dst = srcs[r] + E;

            // h = xin @ W   [WMMA, K=64]
            k_gemm_wmma<64, false, false><<<rowTiles, 128, 0, stream>>>(xin, W, nullptr, h, N);
            k_node_scores<<<(N + 255) / 256, 256, 0, stream>>>(h, as_, ad_, es, ed, mEnc, s, msg, N);
            k_edge_scores<<<(E + 255) / 256, 256, 0, stream>>>(src, dst, es, ed, ebuf, mEnc, E);
            k_edge_expsum<<<(E + 255) / 256, 256, 0, stream>>>(dst, ebuf, mEnc, s, E);
            k_edge_scatter<<<(E + 3) / 4, 256, 0, stream>>>(src, dst, ebuf, h, msg, E);
        }
        k_fuse_ln<<<(N + 7) / 8, 256, 0, stream>>>(xin, msg0, s0, msg1, s1,
                                                   gat_b + (size_t)l * 2 * H,
                                                   gate_w + (size_t)l * H * 2,
                                                   gate_b + l * 2,
                                                   ln_g + l * H, ln_b + l * H, xout, N);
    }

    k_zero64<<<1, 64, 0, stream>>>(sum64);
    k_pool_partial<<<512, 64, 0, stream>>>(out_nodes, sum64, N);
    k_pool_final<<<1, 64, 0, stream>>>(sum64, pool_w, pool_b, out_graph, N);
}